// GATConvWrap_38182259261837
// MI455X (gfx1250) — compile-verified
//
#include <hip/hip_runtime.h>
#include <hip/hip_bf16.h>

// Problem constants (match reference)
#define NN 50000
#define EE 1600000
#define INF 128
#define HH 4
#define FF 32
#define DD 128

typedef __attribute__((ext_vector_type(2))) float v2f;
typedef __attribute__((ext_vector_type(8))) float v8f;
typedef unsigned int u32x4 __attribute__((ext_vector_type(4)));
typedef int i32x8 __attribute__((ext_vector_type(8)));
typedef int i32x4 __attribute__((ext_vector_type(4)));

// ---------------------------------------------------------------------------
// TDM: DMA a rows x 128 f32 tile (row stride 128) from global into LDS,
// with HW padding of 4 DWORDs every 128 DWORDs -> LDS layout [rows][132]
// (bank-conflict-free reads for the WMMA A fragments).
// Descriptor packing per CDNA5 ISA ch.8 (D# groups); 6-arg builtin form
// (clang-23 / therock-10.0 lane).
// ---------------------------------------------------------------------------
__device__ __forceinline__ void tdm_load_tile_f32(const float* gsrc, void* lds_dst,
                                                  int rows) {
    unsigned lds_off = (unsigned)(size_t)lds_dst;            // addr[31:0] = LDS offset
    unsigned long long ga = (unsigned long long)(size_t)gsrc;
    u32x4 g0;
    g0[0] = 1u;                                              // count=1, user descriptor
    g0[1] = lds_off;                                         // lds_addr
    g0[2] = (unsigned)(ga & 0xffffffffu);                    // global_addr[31:0]
    g0[3] = (unsigned)((ga >> 32) & 0x01ffffffu) | (2u << 30); // ga[56:32] | type=2
    i32x8 g1;
    g1[0] = 0x07920000;        // data_size=4B(2<<16) | pad_en(1<<20) | interval=128dw(6<<22) | amount=4dw(3<<25)
    g1[1] = (128 << 16);       // tensor_dim0 = 128 (bits 79:48 low half)
    g1[2] = (rows << 16);      // tensor_dim1 (bits 111:80 low half)
    g1[3] = (128 << 16);       // tile_dim0 = 128 (bits 127:112)
    g1[4] = rows;              // tile_dim1 (bits 143:128)
    g1[5] = 128;               // tensor_dim0_stride (bits 207:160 low 32)
    g1[6] = 0;
    g1[7] = 0;
    i32x4 gz4 = {0, 0, 0, 0};          // 2-D tensor: groups 2/3 unused
    i32x8 gz8 = {0, 0, 0, 0, 0, 0, 0, 0};
    __builtin_amdgcn_tensor_load_to_lds(g0, g1, gz4, gz4, gz8, 0);
}

// ---------------------------------------------------------------------------
// Workspace init: zero accumulators, set segment-max to -inf (ordered-int)
// ---------------------------------------------------------------------------
__global__ __launch_bounds__(256) void init_ws_kernel(int* __restrict__ emax,
                                                      float* __restrict__ denom,
                                                      float* __restrict__ rst,
                                                      float* __restrict__ colsum,
                                                      float* __restrict__ colsq) {
    int t = blockIdx.x * 256 + threadIdx.x;
    if (t < NN * HH) {
        emax[t] = (int)0x80000000;   // INT_MIN == ordered(-inf) lower bound
        denom[t] = 0.0f;
    }
    if (t < NN * DD) rst[t] = 0.0f;
    if (t < DD) { colsum[t] = 0.0f; colsq[t] = 0.0f; }
}

// ---------------------------------------------------------------------------
// GEMM 1: h = x @ W     (50000x128 @ 128x128), fp32 WMMA 16x16x4
// Block = 256 threads = 8 waves; block computes a 16-row x 128-col strip.
// A tile staged via the Tensor Data Mover (wave 0 issues, TENSORcnt sync).
// ---------------------------------------------------------------------------
__global__ __launch_bounds__(256) void gemm_xw_kernel(const float* __restrict__ X,
                                                      const float* __restrict__ W,
                                                      float* __restrict__ Hout) {
    __shared__ float As[16][132];          // padded: stride 132 avoids bank conflicts
    const int row0 = blockIdx.x * 16;
    const int tid = threadIdx.x;

    if ((tid >> 5) == 0) {                 // wave-uniform: TDM ignores EXEC
        tdm_load_tile_f32(X + (size_t)row0 * INF, &As[0][0], 16);
        __builtin_amdgcn_s_wait_tensorcnt(0);
    }
    __syncthreads();

    const int wave = tid >> 5;
    const int lane = tid & 31;
    const int col0 = wave * 16;
    const int m = lane & 15;               // A row within tile / C column index
    const int kp = lane >> 4;              // K-pair selector (0 or 1)
    const int n = m;                       // B/C column within tile

    v8f acc = {};
    for (int k0 = 0; k0 < INF; k0 += 4) {
        v2f a, b;
        a.x = As[m][k0 + 2 * kp];
        a.y = As[m][k0 + 2 * kp + 1];
        b.x = W[(size_t)(k0 + 2 * kp) * DD + col0 + n];
        b.y = W[(size_t)(k0 + 2 * kp + 1) * DD + col0 + n];
        acc = __builtin_amdgcn_wmma_f32_16x16x4_f32(
            /*neg_a=*/false, a, /*neg_b=*/false, b,
            /*c_mod=*/(short)0, acc, /*reuse_a=*/false, /*reuse_b=*/false);
    }

    // C/D layout: VGPR r -> row r (lanes 0-15) or row r+8 (lanes 16-31)
    const int rbase = kp * 8;
    for (int r = 0; r < 8; ++r)
        Hout[(size_t)(row0 + rbase + r) * DD + col0 + n] = acc[r];
}

// ---------------------------------------------------------------------------
// el/er attention dots: el[n,h] = sum_f h[n,h*F+f]*attn_l[h,f]
// ---------------------------------------------------------------------------
__global__ __launch_bounds__(256) void attn_dots_kernel(const float* __restrict__ Hbuf,
                                                        const float* __restrict__ al,
                                                        const float* __restrict__ ar,
                                                        float* __restrict__ el,
                                                        float* __restrict__ er) {
    int t = blockIdx.x * 256 + threadIdx.x;
    if (t >= NN * HH) return;
    int node = t >> 2, hh = t & 3;
    const float* hp = Hbuf + (size_t)node * DD + hh * FF;
    const float* alp = al + hh * FF;
    const float* arp = ar + hh * FF;
    float a = 0.f, b = 0.f;
#pragma unroll
    for (int f = 0; f < FF; ++f) {
        float hv = hp[f];
        a += hv * alp[f];
        b += hv * arp[f];
    }
    el[t] = a;
    er[t] = b;
}

// Monotonic float<->int order-preserving transform (involution per half)
__device__ __forceinline__ int f32_ordered(float v) {
    int b = __float_as_int(v);
    return b >= 0 ? b : (b ^ 0x7fffffff);
}
__device__ __forceinline__ float ordered_f32(int b) {
    return __int_as_float(b >= 0 ? b : (b ^ 0x7fffffff));
}

// ---------------------------------------------------------------------------
// Edge logits + segment max (single-instruction integer atomicMax, no CAS)
// ---------------------------------------------------------------------------
__global__ __launch_bounds__(256) void edge_logits_kernel(const int* __restrict__ src,
                                                          const int* __restrict__ dst,
                                                          const float* __restrict__ el,
                                                          const float* __restrict__ er,
                                                          float* __restrict__ elog,
                                                          int* __restrict__ emax) {
    int t = blockIdx.x * 256 + threadIdx.x;
    if (t >= EE * HH) return;
    int e = t >> 2, hh = t & 3;
    int s = src[e], d = dst[e];
    float v = el[s * HH + hh] + er[d * HH + hh];
    v = v >= 0.f ? v : 0.2f * v;           // leaky_relu(0.2)
    elog[t] = v;
    atomicMax(&emax[d * HH + hh], f32_ordered(v));
}

// ---------------------------------------------------------------------------
// exp(e - max) + segment sum (in place over the logit buffer)
// ---------------------------------------------------------------------------
__global__ __launch_bounds__(256) void edge_exp_kernel(const int* __restrict__ dst,
                                                       float* __restrict__ elog,
                                                       const int* __restrict__ emax,
                                                       float* __restrict__ denom) {
    int t = blockIdx.x * 256 + threadIdx.x;
    if (t >= EE * HH) return;
    int e = t >> 2, hh = t & 3;
    int d = dst[e];
    float mx = ordered_f32(emax[d * HH + hh]);
    float x = __expf(elog[t] - mx);
    elog[t] = x;
    atomicAdd(&denom[d * HH + hh], x);
}

// ---------------------------------------------------------------------------
// Aggregation: rst[dst] += h[src] * alpha.  One wave32 per edge;
// each lane handles 4 contiguous features (float4 gather, 4 f32 atomics).
// Everything (h, rst, denom) is L2-resident on MI455X (192 MB L2).
// ---------------------------------------------------------------------------
__global__ __launch_bounds__(256) void aggregate_kernel(const int* __restrict__ src,
                                                        const int* __restrict__ dst,
                                                        const float* __restrict__ Hbuf,
                                                        const float* __restrict__ ee,
                                                        const float* __restrict__ denom,
                                                        float* __restrict__ rst) {
    const int lane = threadIdx.x & 31;
    const int e = blockIdx.x * 8 + (threadIdx.x >> 5);
    const int s = src[e];
    const int d = dst[e];
    const int head = lane >> 3;            // 8 lanes (32 feats) per head
    const float a = ee[e * HH + head] / denom[d * HH + head];
    const float4 hv = ((const float4*)(Hbuf + (size_t)s * DD))[lane];
    float* rp = rst + (size_t)d * DD + lane * 4;
    atomicAdd(rp + 0, hv.x * a);
    atomicAdd(rp + 1, hv.y * a);
    atomicAdd(rp + 2, hv.z * a);
    atomicAdd(rp + 3, hv.w * a);
}

// ---------------------------------------------------------------------------
// GEMM 2 (+BN stats): out = (rst + bias) @ Wo + bo, fused column sum/sumsq.
// A tile via TDM; bias added in-place in LDS after the DMA lands.
// ---------------------------------------------------------------------------
__global__ __launch_bounds__(256) void gemm_out_kernel(const float* __restrict__ R,
                                                       const float* __restrict__ bias,
                                                       const float* __restrict__ Wo,
                                                       const float* __restrict__ bo,
                                                       float* __restrict__ Out,
                                                       float* __restrict__ colsum,
                                                       float* __restrict__ colsq) {
    __shared__ float As[16][132];
    const int row0 = blockIdx.x * 16;
    const int tid = threadIdx.x;

    if ((tid >> 5) == 0) {
        tdm_load_tile_f32(R + (size_t)row0 * DD, &As[0][0], 16);
        __builtin_amdgcn_s_wait_tensorcnt(0);
    }
    __syncthreads();
    // add GATConv bias to the staged tile (TDM can't fuse the add)
    for (int i = tid; i < 16 * DD; i += 256) {
        int r = i >> 7, c = i & 127;
        As[r][c] += bias[c];
    }
    __syncthreads();

    const int wave = tid >> 5;
    const int lane = tid & 31;
    const int col0 = wave * 16;
    const int m = lane & 15;
    const int kp = lane >> 4;
    const int n = m;

    v8f acc = {};
    for (int k0 = 0; k0 < DD; k0 += 4) {
        v2f a, b;
        a.x = As[m][k0 + 2 * kp];
        a.y = As[m][k0 + 2 * kp + 1];
        b.x = Wo[(size_t)(k0 + 2 * kp) * DD + col0 + n];
        b.y = Wo[(size_t)(k0 + 2 * kp + 1) * DD + col0 + n];
        acc = __builtin_amdgcn_wmma_f32_16x16x4_f32(
            false, a, false, b, (short)0, acc, false, false);
    }

    const float bov = bo[col0 + n];
    const int rbase = kp * 8;
    float s = 0.f, s2 = 0.f;
    for (int r = 0; r < 8; ++r) {
        float val = acc[r] + bov;
        Out[(size_t)(row0 + rbase + r) * DD + col0 + n] = val;
        s += val;
        s2 += val * val;
    }
    // combine the two 8-row halves (lane L and L^16 share the same column n)
    s += __shfl_xor(s, 16);
    s2 += __shfl_xor(s2, 16);
    if (lane < 16) {
        atomicAdd(&colsum[col0 + n], s);
        atomicAdd(&colsq[col0 + n], s2);
    }
}

// ---------------------------------------------------------------------------
// BatchNorm normalize (training stats: mean/var over node axis)
// ---------------------------------------------------------------------------
__global__ __launch_bounds__(256) void bn_norm_kernel(const float* __restrict__ Out,
                                                      const float* __restrict__ colsum,
                                                      const float* __restrict__ colsq,
                                                      const float* __restrict__ gamma,
                                                      const float* __restrict__ beta,
                                                      float* __restrict__ Y) {
    int t = blockIdx.x * 256 + threadIdx.x;
    if (t >= NN * DD) return;
    int c = t & (DD - 1);
    const float invN = 1.0f / (float)NN;
    float mean = colsum[c] * invN;
    float var = colsq[c] * invN - mean * mean;
    Y[t] = (Out[t] - mean) * rsqrtf(var + 1e-5f) * gamma[c] + beta[c];
}

// ---------------------------------------------------------------------------
// Launch
// ---------------------------------------------------------------------------
extern "C" void kernel_launch(void* const* d_in, const int* in_sizes, int n_in,
                              void* d_out, int out_size, void* d_ws, size_t ws_size,
                              hipStream_t stream) {
    const float* x      = (const float*)d_in[0];
    const int*   src    = (const int*)  d_in[1];
    const int*   dst    = (const int*)  d_in[2];
    const float* W      = (const float*)d_in[3];
    const float* attn_l = (const float*)d_in[4];
    const float* attn_r = (const float*)d_in[5];
    const float* bias   = (const float*)d_in[6];
    const float* Wo     = (const float*)d_in[7];
    const float* bo     = (const float*)d_in[8];
    const float* gamma  = (const float*)d_in[9];
    const float* beta   = (const float*)d_in[10];
    float* y = (float*)d_out;

    // Workspace layout (floats)
    float* ws    = (float*)d_ws;
    float* hbuf  = ws;                               // N*D
    float* el    = hbuf + (size_t)NN * DD;           // N*H
    float* er    = el + (size_t)NN * HH;             // N*H
    int*   emax  = (int*)(er + (size_t)NN * HH);     // N*H (ordered ints)
    float* denom = (float*)(emax + (size_t)NN * HH); // N*H
    float* elog  = denom + (size_t)NN * HH;          // E*H (logits -> exp in place)
    float* rst   = elog + (size_t)EE * HH;           // N*D
    float* outb  = rst + (size_t)NN * DD;            // N*D
    float* csum  = outb + (size_t)NN * DD;           // D
    float* csq   = csum + DD;                        // D

    const int ndBlocks = (NN * DD + 255) / 256;      // 25000
    const int ehBlocks = (EE * HH + 255) / 256;      // 25000
    const int nhBlocks = (NN * HH + 255) / 256;      // 782
    const int gemmBlocks = NN / 16;                  // 3125
    const int aggBlocks = EE / 8;                    // 200000

    init_ws_kernel<<<ndBlocks, 256, 0, stream>>>(emax, denom, rst, csum, csq);
    gemm_xw_kernel<<<gemmBlocks, 256, 0, stream>>>(x, W, hbuf);
    attn_dots_kernel<<<nhBlocks, 256, 0, stream>>>(hbuf, attn_l, attn_r, el, er);
    edge_logits_kernel<<<ehBlocks, 256, 0, stream>>>(src, dst, el, er, elog, emax);
    edge_exp_kernel<<<ehBlocks, 256, 0, stream>>>(dst, elog, emax, denom);
    aggregate_kernel<<<aggBlocks, 256, 0, stream>>>(src, dst, hbuf, elog, denom, rst);
    gemm_out_kernel<<<gemmBlocks, 256, 0, stream>>>(rst, bias, Wo, bo, outb, csum, csq);
    bn_norm_kernel<<<ndBlocks, 256, 0, stream>>>(outb, csum, csq, gamma, beta, y);
}